// DAB_46213848105535
// MI455X (gfx1250) — compile-verified
//
#include <hip/hip_runtime.h>

typedef __attribute__((ext_vector_type(2))) float v2f;
typedef __attribute__((ext_vector_type(8))) float v8f;
typedef unsigned int u32x4 __attribute__((ext_vector_type(4)));
typedef int i32x4 __attribute__((ext_vector_type(4)));
typedef int i32x8 __attribute__((ext_vector_type(8)));

#define HH 128
#define WW 128
#define CC 64
#define HWSZ (HH * WW)
#define NN 8
#define KT 9          // 3x3 taps
#define OFFC 18       // 2*K offset channels
#define BN_EPS 1e-5f

// ---------------------------------------------------------------------------
// K0: offset conv 3x3, 64 -> 18 channels, pad 1, + bias.
// One block per (n, h) row; one thread per pixel (w). Weight indices are
// thread-uniform -> scalar (s_load) broadcasts.
// ---------------------------------------------------------------------------
__global__ __launch_bounds__(128) void dab_offset_conv(
    const float* __restrict__ x, const float* __restrict__ w_off,
    const float* __restrict__ b_off, float* __restrict__ offs) {
  const int h = blockIdx.x % HH;
  const int n = blockIdx.x / HH;
  const int px = threadIdx.x;

  float acc[OFFC];
#pragma unroll
  for (int oc = 0; oc < OFFC; ++oc) acc[oc] = b_off[oc];

  const float* xn = x + (size_t)n * CC * HWSZ;
  for (int c = 0; c < CC; ++c) {
    const float* xp = xn + c * HWSZ;
#pragma unroll
    for (int t = 0; t < KT; ++t) {
      const int ky = t / 3, kx = t - 3 * (t / 3);
      const int y = h + ky - 1;
      const int xq = px + kx - 1;
      float xv = 0.0f;
      if (y >= 0 && y < HH && xq >= 0 && xq < WW) xv = xp[y * WW + xq];
#pragma unroll
      for (int oc = 0; oc < OFFC; ++oc)
        acc[oc] = fmaf(xv, w_off[(oc * CC + c) * KT + t], acc[oc]);
    }
  }
  float* op = offs + (size_t)n * OFFC * HWSZ + h * WW + px;
#pragma unroll
  for (int oc = 0; oc < OFFC; ++oc) op[(size_t)oc * HWSZ] = acc[oc];
}

// ---------------------------------------------------------------------------
// K0b: one-time weight transpose w[o][c][k] -> wt[k][c][o] so each tap's
// 64x64 slice is a contiguous 16 KB block (TDM-friendly 1D tile).
// ---------------------------------------------------------------------------
__global__ __launch_bounds__(256) void dab_weight_reorder(
    const float* __restrict__ w, float* __restrict__ wt) {
  const int idx = blockIdx.x * 256 + threadIdx.x;  // KT*CC*CC = 36864 total
  if (idx < KT * CC * CC) {
    const int k = idx >> 12;       // /4096
    const int r = idx & 4095;
    const int c = r >> 6, o = r & 63;
    wt[idx] = w[(o * CC + c) * KT + k];
  }
}

// ---------------------------------------------------------------------------
// TDM: issue one tensor_load_to_lds DMA of a contiguous 4096-float (16 KB)
// slice into LDS. Descriptor per CDNA5 ISA 8.3/8.4:
//   group0: count=1 | lds_addr[63:32] | global_addr[120:64] | type=2[127:126]
//   group1: data_size=4B[17:16] | tensor_dim0=4096[79:48] | tensor_dim1=1
//           | tile_dim0=4096[127:112] | tile_dim1=1 | dim0_stride=4096
//   groups 2/3: zero (<=2D tile)
// ---------------------------------------------------------------------------
__device__ __forceinline__ void tdm_load_16kb(uint32_t lds_addr,
                                              const float* gsrc) {
  const uint64_t ga = (uint64_t)(uintptr_t)gsrc;
  u32x4 g0;
  g0[0] = 1u;                                   // count=1 (valid user D#)
  g0[1] = lds_addr;                             // LDS byte address
  g0[2] = (uint32_t)(ga & 0xFFFFFFFFu);         // global_addr[95:64]
  g0[3] = (uint32_t)((ga >> 32) & 0x01FFFFFFu)  // global_addr[120:96]
          | (2u << 30);                         // type=2 ("image")
  i32x8 g1;
  g1[0] = 0x00020000;                           // data_size=2 (4 bytes)
  g1[1] = (int)(0x1000u << 16);                 // tensor_dim0 = 4096 (lo16)
  g1[2] = (int)(1u << 16);                      // tensor_dim1 = 1
  g1[3] = (int)(0x1000u << 16);                 // tile_dim0 = 4096
  g1[4] = 1;                                    // tile_dim1 = 1, tile_dim2 = 0
  g1[5] = 0x1000;                               // tensor_dim0_stride = 4096
  g1[6] = 0;
  g1[7] = 0;
  const i32x4 gz = {};
#if __clang_major__ >= 23
  const i32x8 gz8 = {};
  __builtin_amdgcn_tensor_load_to_lds(g0, g1, gz, gz, gz8, 0);
#else
  __builtin_amdgcn_tensor_load_to_lds(g0, g1, gz, gz, 0);
#endif
}

// ---------------------------------------------------------------------------
// K1: fused bilinear gather + deformable GEMM via V_WMMA_F32_16X16X4_F32.
// Block = 256 threads (8 waves) handles one (n, h) row of 128 pixels.
// Wave w owns pixels [w*16, w*16+16) and all 64 output channels
// (4 accumulator tiles of 16x16). Reduction dim reordered as ck' = k*64 + c
// so bilinear coords hoist out of the channel loop. Per-tap weight slices
// arrive via double-buffered TDM DMA overlapped with compute.
//
// A frag (16x4 f32, MxK): lane<16 -> M=lane, a.x=K0, a.y=K1;
//                         lane>=16 -> M=lane-16, a.x=K2, a.y=K3.
// B frag (4x16 f32, KxN): mirrored with N on lanes.
// ---------------------------------------------------------------------------
__global__ __launch_bounds__(256) void dab_deform_wmma(
    const float* __restrict__ x, const float* __restrict__ offs,
    const float* __restrict__ wt, const float* __restrict__ bias,
    float* __restrict__ out_pre) {
  __shared__ unsigned short sIdx[WW * KT][4];  // clamped corner indices (<16384)
  __shared__ float sWgt[WW * KT][4];           // bilinear wgts, validity folded
  __shared__ float sWk[2][CC][CC];             // double-buffered W[:, :, k] [c][o]

  const int tid = threadIdx.x;
  const int n = blockIdx.x / HH;
  const int h = blockIdx.x % HH;
  const int wave = tid >> 5;
  const int lane = tid & 31;
  const int half = lane >> 4;
  const int ml = lane & 15;
  const int px = wave * 16 + ml;

  // kick off tap-0 weight DMA immediately (no prior readers of buf 0)
  if (wave == 0) {
    tdm_load_16kb((uint32_t)(uintptr_t)&sWk[0][0][0], wt);
  }

  // --- setup: per-(pixel, tap) bilinear coordinates ---
  const float* offn = offs + (size_t)n * OFFC * HWSZ;
  for (int e = tid; e < WW * KT; e += 256) {
    const int ppx = e / KT;
    const int k = e - ppx * KT;
    const int ky = k / 3, kx = k - 3 * (k / 3);
    const float dy = offn[(size_t)(2 * k) * HWSZ + h * WW + ppx];
    const float dx = offn[(size_t)(2 * k + 1) * HWSZ + h * WW + ppx];
    const float ys = (float)(h + ky - 1) + dy;
    const float xs = (float)(ppx + kx - 1) + dx;
    const float y0f = floorf(ys), x0f = floorf(xs);
    const float wy = ys - y0f, wx = xs - x0f;
    const int iy0 = (int)y0f, ix0 = (int)x0f;
    const int iy1 = iy0 + 1, ix1 = ix0 + 1;
    const float vy0 = (iy0 >= 0 && iy0 < HH) ? 1.0f : 0.0f;
    const float vy1 = (iy1 >= 0 && iy1 < HH) ? 1.0f : 0.0f;
    const float vx0 = (ix0 >= 0 && ix0 < WW) ? 1.0f : 0.0f;
    const float vx1 = (ix1 >= 0 && ix1 < WW) ? 1.0f : 0.0f;
    const int cy0 = min(max(iy0, 0), HH - 1), cy1 = min(max(iy1, 0), HH - 1);
    const int cx0 = min(max(ix0, 0), WW - 1), cx1 = min(max(ix1, 0), WW - 1);
    sIdx[e][0] = (unsigned short)(cy0 * WW + cx0);
    sIdx[e][1] = (unsigned short)(cy0 * WW + cx1);
    sIdx[e][2] = (unsigned short)(cy1 * WW + cx0);
    sIdx[e][3] = (unsigned short)(cy1 * WW + cx1);
    sWgt[e][0] = (1.0f - wy) * (1.0f - wx) * vy0 * vx0;
    sWgt[e][1] = (1.0f - wy) * wx * vy0 * vx1;
    sWgt[e][2] = wy * (1.0f - wx) * vy1 * vx0;
    sWgt[e][3] = wy * wx * vy1 * vx1;
  }

  const float* xn = x + (size_t)n * CC * HWSZ;
  v8f acc0 = {}, acc1 = {}, acc2 = {}, acc3 = {};

  for (int k = 0; k < KT; ++k) {
    if (wave == 0) __builtin_amdgcn_s_wait_tensorcnt(0);  // weight slice k landed
    __syncthreads();  // publish sWk[k&1] (+ coord setup on k==0); all reads of
                      // parity (k+1)&1 (from iter k-1) completed before here
    if (wave == 0 && k + 1 < KT) {
      tdm_load_16kb((uint32_t)(uintptr_t)&sWk[(k + 1) & 1][0][0],
                    wt + (size_t)(k + 1) * (CC * CC));  // overlaps compute
    }
    const float(*Wk)[CC] = sWk[k & 1];

    // hoist this lane's bilinear coords for tap k (constant across channels)
    const int e = px * KT + k;
    const int i0 = sIdx[e][0], i1 = sIdx[e][1], i2 = sIdx[e][2], i3 = sIdx[e][3];
    const float w0 = sWgt[e][0], w1 = sWgt[e][1], w2 = sWgt[e][2], w3 = sWgt[e][3];

    for (int c0 = 0; c0 < CC; c0 += 4) {
      const int ca = c0 + half * 2;  // channel for a.x / b.x row
      const float* pa = xn + (size_t)ca * HWSZ;
      const float* pb = pa + HWSZ;
      v2f a;
      a.x = w0 * pa[i0] + w1 * pa[i1] + w2 * pa[i2] + w3 * pa[i3];
      a.y = w0 * pb[i0] + w1 * pb[i1] + w2 * pb[i2] + w3 * pb[i3];

      v2f b0, b1, b2, b3;
      b0.x = Wk[ca][ml];        b0.y = Wk[ca + 1][ml];
      b1.x = Wk[ca][16 + ml];   b1.y = Wk[ca + 1][16 + ml];
      b2.x = Wk[ca][32 + ml];   b2.y = Wk[ca + 1][32 + ml];
      b3.x = Wk[ca][48 + ml];   b3.y = Wk[ca + 1][48 + ml];

      acc0 = __builtin_amdgcn_wmma_f32_16x16x4_f32(false, a, false, b0, (short)0, acc0, false, false);
      acc1 = __builtin_amdgcn_wmma_f32_16x16x4_f32(false, a, false, b1, (short)0, acc1, false, false);
      acc2 = __builtin_amdgcn_wmma_f32_16x16x4_f32(false, a, false, b2, (short)0, acc2, false, false);
      acc3 = __builtin_amdgcn_wmma_f32_16x16x4_f32(false, a, false, b3, (short)0, acc3, false, false);
    }
  }

  // --- store + bias. C/D layout: vgpr r, lane<16 -> (M=r, N=lane);
  //     lane>=16 -> (M=r+8, N=lane-16). ---
  const float bo0 = bias[ml];
  const float bo1 = bias[16 + ml];
  const float bo2 = bias[32 + ml];
  const float bo3 = bias[48 + ml];
  float* outn = out_pre + (size_t)n * CC * HWSZ + h * WW;
#pragma unroll
  for (int r = 0; r < 8; ++r) {
    const int pxo = wave * 16 + r + 8 * half;
    outn[(size_t)(ml)*HWSZ + pxo]        = acc0[r] + bo0;
    outn[(size_t)(16 + ml) * HWSZ + pxo] = acc1[r] + bo1;
    outn[(size_t)(32 + ml) * HWSZ + pxo] = acc2[r] + bo2;
    outn[(size_t)(48 + ml) * HWSZ + pxo] = acc3[r] + bo3;
  }
}

// ---------------------------------------------------------------------------
// K2a: zero the stats buffer (must re-init every launch).
// ---------------------------------------------------------------------------
__global__ void dab_zero_stats(float* __restrict__ stats) {
  if (threadIdx.x < 128) stats[threadIdx.x] = 0.0f;
}

// ---------------------------------------------------------------------------
// K2b: per-channel sum / sum-of-squares (block per (n, channel), atomics).
// ---------------------------------------------------------------------------
__global__ __launch_bounds__(256) void dab_bn_stats(
    const float* __restrict__ pre, float* __restrict__ stats) {
  __shared__ float s1[256], s2[256];
  const int o = blockIdx.x & 63;
  const int n = blockIdx.x >> 6;
  const float* p = pre + ((size_t)n * CC + o) * HWSZ;
  float s = 0.0f, q = 0.0f;
  for (int i = threadIdx.x; i < HWSZ; i += 256) {
    const float v = p[i];
    s += v;
    q += v * v;
  }
  s1[threadIdx.x] = s;
  s2[threadIdx.x] = q;
  __syncthreads();
  for (int st = 128; st > 0; st >>= 1) {
    if (threadIdx.x < st) {
      s1[threadIdx.x] += s1[threadIdx.x + st];
      s2[threadIdx.x] += s2[threadIdx.x + st];
    }
    __syncthreads();
  }
  if (threadIdx.x == 0) {
    atomicAdd(&stats[o], s1[0]);
    atomicAdd(&stats[64 + o], s2[0]);
  }
}

// ---------------------------------------------------------------------------
// K3: BN finalize: out = (pre - mean) * rsqrt(var + eps) * gamma + beta
// ---------------------------------------------------------------------------
__global__ __launch_bounds__(256) void dab_bn_apply(
    const float* __restrict__ pre, const float* __restrict__ stats,
    const float* __restrict__ gamma, const float* __restrict__ beta,
    float* __restrict__ out, long long total) {
  const float inv_cnt = 1.0f / (float)(NN * HWSZ);
  for (long long i = (long long)blockIdx.x * 256 + threadIdx.x; i < total;
       i += (long long)gridDim.x * 256) {
    const int o = (int)((i / HWSZ) & 63);
    const float mean = stats[o] * inv_cnt;
    const float var = stats[64 + o] * inv_cnt - mean * mean;
    const float sc = gamma[o] * rsqrtf(var + BN_EPS);
    out[i] = (pre[i] - mean) * sc + beta[o];
  }
}

extern "C" void kernel_launch(void* const* d_in, const int* in_sizes, int n_in,
                              void* d_out, int out_size, void* d_ws, size_t ws_size,
                              hipStream_t stream) {
  (void)in_sizes; (void)n_in; (void)ws_size;
  const float* x     = (const float*)d_in[0];
  const float* w_off = (const float*)d_in[1];
  const float* b_off = (const float*)d_in[2];
  const float* w     = (const float*)d_in[3];
  const float* b     = (const float*)d_in[4];
  const float* gamma = (const float*)d_in[5];
  const float* beta  = (const float*)d_in[6];
  float* out = (float*)d_out;

  // workspace layout (floats): offsets | pre-BN output | stats | wt
  float* offs  = (float*)d_ws;                              // 8*18*128*128
  float* pre   = offs + (size_t)NN * OFFC * HWSZ;           // 8*64*128*128
  float* stats = pre + (size_t)NN * CC * HWSZ;              // 128
  float* wt    = stats + 128;                               // 9*64*64

  dab_offset_conv<<<NN * HH, 128, 0, stream>>>(x, w_off, b_off, offs);
  dab_weight_reorder<<<(KT * CC * CC + 255) / 256, 256, 0, stream>>>(w, wt);
  dab_zero_stats<<<1, 128, 0, stream>>>(stats);
  dab_deform_wmma<<<NN * HH, 256, 0, stream>>>(x, offs, wt, b, pre);
  dab_bn_stats<<<NN * CC, 256, 0, stream>>>(pre, stats);
  dab_bn_apply<<<4096, 256, 0, stream>>>(pre, stats, gamma, beta, out,
                                         (long long)out_size);
}